// DiversityLoss_88776974008411
// MI455X (gfx1250) — compile-verified
//
#include <hip/hip_runtime.h>
#include <hip/hip_bf16.h>

// ---------------------------------------------------------------------------
// DiversityLoss: per-class mean of 1M x 128 f32 embeddings over 1000 classes,
// then -mean(var(class_means, ddof=1)).
// Memory-bound: 512MB stream @ 23.3TB/s ~ 22us. Strategy:
//   K0: zero sums[1000x128] + counts[1000] in d_ws
//   K1: LDS-aggregated segment sum (ds_add_f32), flush with few global atomics
//   K2: finalize with f32 WMMA (ones-vector x means matrix column sums)
// ---------------------------------------------------------------------------

#define NUM_C   1000
#define DIM     128
#define GD      32          // dims per group (4 groups cover 128)
#define CSTRIDE 33          // padded LDS stride per class (bank-conflict break)
#define NCHUNK  128         // row chunks -> 512 blocks total

typedef float v2f __attribute__((ext_vector_type(2)));
typedef float v8f __attribute__((ext_vector_type(8)));

__global__ void dl_zero_kernel(float* __restrict__ p, int n) {
    int i = blockIdx.x * 256 + threadIdx.x;
    if (i < n) p[i] = 0.0f;
}

// Each block: (row chunk, 32-dim group). LDS table lsum[1000][33] + lcnt[1000].
// 8 lanes per row load a float4 each (128B contiguous per row slice).
__global__ __launch_bounds__(256) void dl_segsum_kernel(
    const float* __restrict__ emb, const int* __restrict__ labels,
    float* __restrict__ sums, float* __restrict__ counts, int N) {
    extern __shared__ float smem[];
    float* lsum = smem;                       // NUM_C * CSTRIDE
    float* lcnt = smem + NUM_C * CSTRIDE;     // NUM_C

    const int g     = blockIdx.x & 3;
    const int chunk = blockIdx.x >> 2;
    const int tid   = threadIdx.x;

    for (int i = tid; i < NUM_C * CSTRIDE; i += 256) lsum[i] = 0.0f;
    for (int i = tid; i < NUM_C; i += 256) lcnt[i] = 0.0f;
    __syncthreads();

    const int R     = (N + NCHUNK - 1) / NCHUNK;
    const int start = chunk * R;
    const int end   = (start + R < N) ? (start + R) : N;
    const int sub   = tid & 7;                // 8 threads per row
    const int base_d = g * GD + sub * 4;

    for (int r = start + (tid >> 3); r < end; r += 32) {
        const float4 v =
            *reinterpret_cast<const float4*>(emb + (size_t)r * DIM + base_d);
        const int lab = labels[r];            // broadcast within the 8 lanes
        float* p = lsum + lab * CSTRIDE + sub * 4;
        atomicAdd(p + 0, v.x);
        atomicAdd(p + 1, v.y);
        atomicAdd(p + 2, v.z);
        atomicAdd(p + 3, v.w);
        if (g == 0 && sub == 0) atomicAdd(lcnt + lab, 1.0f);
    }
    __syncthreads();

    // Flush partial sums: 32K global atomics per block (~16M total).
    for (int i = tid; i < NUM_C * GD; i += 256) {
        const int c = i >> 5;
        const int d = i & 31;
        atomicAdd(sums + c * DIM + g * GD + d, lsum[c * CSTRIDE + d]);
    }
    if (g == 0)
        for (int i = tid; i < NUM_C; i += 256) atomicAdd(counts + i, lcnt[i]);
}

// Finalize: one block, 8 waves. Wave w owns dims [16w, 16w+16).
// Column sums over classes of M and M^2 via f32 WMMA with all-ones A:
//   D[m][n] += sum_k A[m][k]*B[k][n], A==1 => every row of D = colsum(B).
// Exact f32 arithmetic; invariant to K striping since A is uniform.
__global__ __launch_bounds__(256) void dl_finalize_kernel(
    const float* __restrict__ sums, const float* __restrict__ counts,
    float* __restrict__ out) {
    __shared__ float vvar[DIM];
    const int tid  = threadIdx.x;
    const int wave = tid >> 5;
    const int lane = tid & 31;
    const int d0   = wave * 16;
    const int n    = lane & 15;
    const int kh   = lane >> 4;               // lanes 16-31 hold K=2,3

    v8f S = {};
    v8f Q = {};
    v2f A;
    A.x = 1.0f; A.y = 1.0f;

    for (int c0 = 0; c0 < NUM_C; c0 += 4) {
        const int c = c0 + 2 * kh;
        const float m0 = sums[(size_t)c * DIM + d0 + n] / counts[c];
        const float m1 = sums[(size_t)(c + 1) * DIM + d0 + n] / counts[c + 1];
        v2f B;  B.x  = m0;       B.y  = m1;
        v2f B2; B2.x = m0 * m0;  B2.y = m1 * m1;
        S = __builtin_amdgcn_wmma_f32_16x16x4_f32(false, A, false, B,
                                                  (short)0, S, false, false);
        Q = __builtin_amdgcn_wmma_f32_16x16x4_f32(false, A, false, B2,
                                                  (short)0, Q, false, false);
    }

    if (lane < 16) {
        const float s = S[0];                 // row m=0: colsum of means
        const float q = Q[0];                 // colsum of means^2
        const float C = (float)NUM_C;
        vvar[d0 + n] = (q - s * s / C) / (C - 1.0f);   // ddof=1
    }
    __syncthreads();
    if (tid == 0) {
        float acc = 0.0f;
        for (int d = 0; d < DIM; ++d) acc += vvar[d];
        out[0] = -(acc / (float)DIM);
    }
}

extern "C" void kernel_launch(void* const* d_in, const int* in_sizes, int n_in,
                              void* d_out, int out_size, void* d_ws, size_t ws_size,
                              hipStream_t stream) {
    (void)n_in; (void)out_size; (void)ws_size;
    const float* emb   = (const float*)d_in[0];
    const int* labels  = (const int*)d_in[1];
    const int N        = in_sizes[1];         // number of rows / labels

    float* sums   = (float*)d_ws;             // [1000][128]
    float* counts = sums + NUM_C * DIM;       // [1000] (contiguous after sums)
    float* out    = (float*)d_out;

    const int tot = NUM_C * DIM + NUM_C;
    dl_zero_kernel<<<(tot + 255) / 256, 256, 0, stream>>>(sums, tot);

    const size_t lds_bytes = (size_t)(NUM_C * CSTRIDE + NUM_C) * sizeof(float);
    (void)hipFuncSetAttribute((const void*)dl_segsum_kernel,
                              hipFuncAttributeMaxDynamicSharedMemorySize,
                              (int)lds_bytes);
    dl_segsum_kernel<<<NCHUNK * 4, 256, lds_bytes, stream>>>(emb, labels, sums,
                                                             counts, N);

    dl_finalize_kernel<<<1, 256, 0, stream>>>(sums, counts, out);
}